// LangevinSDEContiformer_22582938042495
// MI455X (gfx1250) — compile-verified
//
#include <hip/hip_runtime.h>
#include <cmath>

// ---------------------------------------------------------------------------
// Langevin SDE scan for MI455X (gfx1250, wave32).
//  * fp32 end-to-end on V_WMMA_F32_16X16X4_F32 (exact fp32 matrix path)
//  * one wave == one 16-row batch tile, loops all 511 steps
//  * weights pre-swizzled in LDS to B-fragment order: B fragments arrive as
//    conflict-free ds_load_2addr_b64 with immediate offsets
//  * 4 accumulators share each A fragment -> 4 interleaved WMMA chains
//  * gfx1250 hardware V_TANH_F32 for the activation epilogues
// ---------------------------------------------------------------------------

typedef __attribute__((ext_vector_type(2))) float v2f;
typedef __attribute__((ext_vector_type(8))) float v8f;

#define B_SZ   1024
#define S_SZ   512
#define H_SZ   64
#define DIN    3
#define NSTEP  (S_SZ - 1)   // 511
#define H2     128          // 2*H
#define HP     68           // H+1 = 65 padded to multiple of 4 (zero-filled)

__device__ __forceinline__ float fast_tanh(float x) {
#if __has_builtin(__builtin_amdgcn_tanhf)
  return __builtin_amdgcn_tanhf(x);     // single V_TANH_F32 (TRANS pipe)
#else
  return tanhf(x);
#endif
}

// A fragment (16x4 f32, ISA 7.12.2): lanes 0-15 -> M=0..15,K={0,1};
// lanes 16-31 -> same M, K={2,3}.  One ds_load_b64 from a row-major strip.
__device__ __forceinline__ v2f ldA(const float* buf, int stride, int kt,
                                   int ml, int mh) {
  const float* p = buf + ml * stride + 4 * kt + (mh << 1);
  v2f a; a.x = p[0]; a.y = p[1]; return a;
}

// B fragment from a pre-swizzled image: frag (kt,nt) is 64 consecutive floats,
// lane's pair at +lane*2.  Address folds to (base + lane*8) + const offset.
__device__ __forceinline__ v2f ldBs(const float* img, int KT, int nt, int kt,
                                    int lane) {
  return ((const v2f*)img)[(nt * KT + kt) * 32 + lane];
}

__device__ __forceinline__ v8f wmma4(v2f a, v2f b, v8f c) {
  return __builtin_amdgcn_wmma_f32_16x16x4_f32(false, a, false, b,
                                               (short)0, c, false, false);
}

// Build a B-swizzled image of a logical KxN matrix (K = 4*KT).
// element (k,n) -> ((n/16)*KT + k/4)*64 + (((k>>1)&1)*16 + (n&15))*2 + (k&1)
template <typename SRC>
__device__ __forceinline__ void buildB(float* img, int KT, int N, int lane,
                                       SRC src) {
  const int K = KT * 4;
  for (int i = lane; i < K * N; i += 32) {
    int k = i / N, n = i % N;
    int dst = ((n >> 4) * KT + (k >> 2)) * 64 +
              (((k >> 1) & 1) * 16 + (n & 15)) * 2 + (k & 1);
    img[dst] = src(k, n);
  }
}

__global__ void __launch_bounds__(32)
sde_scan_kernel(const float* __restrict__ ts,     // (B,S,3)
                const float* __restrict__ noise,  // (511,B,64)
                const float* __restrict__ Wp1,    // (65,128)
                const float* __restrict__ bp1,    // (128)
                const float* __restrict__ Wp2,    // (128,64)
                const float* __restrict__ bp2,    // (64)
                const float* __restrict__ Wp3,    // (64)
                const float* __restrict__ Wd1,    // (32)
                const float* __restrict__ bd1,    // (32)
                const float* __restrict__ Wd2,    // (32,64)
                const float* __restrict__ bd2,    // (64)
                const float* __restrict__ mind_p,
                const float* __restrict__ maxd_p,
                float* __restrict__ out)          // (B,S,64)
{
  extern __shared__ float sm[];
  float* sF1  = sm;                       // 17*8*64  Wp1   (fwd1, K=68,N=128)
  float* sF2  = sF1 + 17 * 8 * 64;        // 32*4*64  Wp2   (fwd2, K=128,N=64)
  float* sB1  = sF2 + 32 * 4 * 64;        // 16*8*64  Wp2^T (bwd1, K=64,N=128)
  float* sB2  = sB1 + 16 * 8 * 64;        // 32*4*64  Wp1^T (bwd2, K=128,N=64)
  float* sBp1 = sB2 + 32 * 4 * 64;        // 128
  float* sBp2 = sBp1 + H2;                // 64
  float* sWp3 = sBp2 + H_SZ;              // 64
  float* sT   = sWp3 + H_SZ;              // 512
  float* sSig = sT + S_SZ;                // 511*64: sigma(t_s)[j]*sqrt(h_s)
  float* sTy  = sSig + NSTEP * H_SZ;      // 16*HP  [y | t | 0pad]
  float* sH1  = sTy + 16 * HP;            // 16*128
  float* sDb  = sH1 + 16 * H2;            // 16*64   dh2_pre
  float* sD2  = sDb + 16 * H_SZ;          // 16*128  dh1_pre

  const int lane = threadIdx.x;      // wave32
  const int ml   = lane & 15;
  const int mh   = lane >> 4;
  const int m0   = blockIdx.x * 16;  // 64 blocks cover B=1024

  // ---- one-time: swizzled weight images + scalars into LDS -----------------
  buildB(sF1, 17, 128, lane, [&](int k, int n) {
    return (k < 65) ? Wp1[k * H2 + n] : 0.0f; });
  buildB(sF2, 32, 64, lane, [&](int k, int n) {
    return Wp2[k * H_SZ + n]; });
  buildB(sB1, 16, 128, lane, [&](int k, int n) {   // Wp2^T
    return Wp2[n * H_SZ + k]; });
  buildB(sB2, 32, 64, lane, [&](int k, int n) {    // Wp1^T (first 64 rows)
    return Wp1[n * H2 + k]; });
  for (int i = lane; i < H2; i += 32)  sBp1[i] = bp1[i];
  for (int i = lane; i < H_SZ; i += 32) { sBp2[i] = bp2[i]; sWp3[i] = Wp3[i]; }
  for (int i = lane; i < S_SZ; i += 32) sT[i] = ts[i * DIN];
  for (int r = lane; r < 16; r += 32) {            // zero ty K-pad columns
    sTy[r * HP + 65] = 0.f; sTy[r * HP + 66] = 0.f; sTy[r * HP + 67] = 0.f;
  }
  __syncthreads();

  // ---- one-time: sigma(t_s)*sqrt(h_s) table --------------------------------
  const float mind = fabsf(mind_p[0]);
  const float maxd = fabsf(maxd_p[0]);
#pragma unroll 1
  for (int e = lane; e < NSTEP * H_SZ; e += 32) {
    int s = e >> 6, j = e & 63;
    float t0 = sT[s];
    float acc = bd2[j];
#pragma unroll
    for (int i = 0; i < 32; ++i) {
      float hh = t0 * Wd1[i] + bd1[i];
      hh = hh > 0.f ? hh : 0.f;
      acc += hh * Wd2[i * H_SZ + j];
    }
    float sp = (acc > 20.f) ? acc : log1pf(__expf(acc));
    float sg = fminf(fmaxf(sp + mind, mind), maxd);
    sSig[e] = sg * sqrtf(sT[s + 1] - t0);
  }
  __syncthreads();

  // ---- y0 in WMMA C-layout registers; emit trajectory row 0 ----------------
  v8f y[4];
#pragma unroll
  for (int nt = 0; nt < 4; ++nt) {
#pragma unroll
    for (int r = 0; r < 8; ++r) {
      int m = m0 + r + mh * 8;
      int j = nt * 16 + ml;
      float v = (j < DIN) ? ts[(size_t)m * S_SZ * DIN + j] : 0.0f;
      y[nt][r] = v;
      out[(size_t)m * S_SZ * H_SZ + j] = v;
    }
  }

  // ---- main scan -----------------------------------------------------------
#pragma unroll 1
  for (int s = 0; s < NSTEP; ++s) {
    const float t0 = sT[s];
    const float h  = sT[s + 1] - t0;

    if (s + 1 < NSTEP) {   // prefetch next step's contiguous 4KB noise tile
      const char* np = (const char*)(noise + (size_t)(s + 1) * B_SZ * H_SZ
                                           + (size_t)m0 * H_SZ);
      __builtin_prefetch(np + lane * 128, 0, 1);
    }

    // stage ty = [y | t0 | 0pad]
#pragma unroll
    for (int nt = 0; nt < 4; ++nt)
#pragma unroll
      for (int r = 0; r < 8; ++r)
        sTy[(r + mh * 8) * HP + nt * 16 + ml] = y[nt][r];
    if (lane < 16) sTy[lane * HP + 64] = t0;
    __syncthreads();

    // fwd1: h1 = tanh(ty @ Wp1 + bp1)   2 groups x 4 shared-A chains
#pragma unroll
    for (int g = 0; g < 2; ++g) {
      v8f c[4];
#pragma unroll
      for (int t = 0; t < 4; ++t) {
        float b = sBp1[(4 * g + t) * 16 + ml];
        c[t] = (v8f){b, b, b, b, b, b, b, b};
      }
#pragma unroll
      for (int kt = 0; kt < 17; ++kt) {
        v2f a = ldA(sTy, HP, kt, ml, mh);
#pragma unroll
        for (int t = 0; t < 4; ++t)
          c[t] = wmma4(a, ldBs(sF1, 17, 4 * g + t, kt, lane), c[t]);
      }
#pragma unroll
      for (int t = 0; t < 4; ++t)
#pragma unroll
        for (int r = 0; r < 8; ++r)
          sH1[(r + mh * 8) * H2 + (4 * g + t) * 16 + ml] = fast_tanh(c[t][r]);
    }
    __syncthreads();

    // fwd2: h2 = tanh(h1@Wp2+bp2); dh2_pre = Wp3 * (1 - h2^2)
    {
      v8f c[4];
#pragma unroll
      for (int t = 0; t < 4; ++t) {
        float b = sBp2[t * 16 + ml];
        c[t] = (v8f){b, b, b, b, b, b, b, b};
      }
#pragma unroll
      for (int kt = 0; kt < 32; ++kt) {
        v2f a = ldA(sH1, H2, kt, ml, mh);
#pragma unroll
        for (int t = 0; t < 4; ++t)
          c[t] = wmma4(a, ldBs(sF2, 32, t, kt, lane), c[t]);
      }
#pragma unroll
      for (int t = 0; t < 4; ++t) {
        float g3 = sWp3[t * 16 + ml];
#pragma unroll
        for (int r = 0; r < 8; ++r) {
          float t2 = fast_tanh(c[t][r]);
          sDb[(r + mh * 8) * H_SZ + t * 16 + ml] = g3 * (1.0f - t2 * t2);
        }
      }
    }
    __syncthreads();

    // bwd1: dh1_pre = (dh2_pre @ Wp2^T) * (1 - h1^2)
#pragma unroll
    for (int g = 0; g < 2; ++g) {
      v8f c[4] = {};
#pragma unroll
      for (int kt = 0; kt < 16; ++kt) {
        v2f a = ldA(sDb, H_SZ, kt, ml, mh);
#pragma unroll
        for (int t = 0; t < 4; ++t)
          c[t] = wmma4(a, ldBs(sB1, 16, 4 * g + t, kt, lane), c[t]);
      }
#pragma unroll
      for (int t = 0; t < 4; ++t)
#pragma unroll
        for (int r = 0; r < 8; ++r) {
          int idx = (r + mh * 8) * H2 + (4 * g + t) * 16 + ml;
          float hv = sH1[idx];                       // re-read h1 (reg relief)
          sD2[idx] = c[t][r] * (1.0f - hv * hv);
        }
    }
    __syncthreads();

    // bwd2: dty = dh1_pre @ Wp1^T ; Euler-Maruyama update + store
    {
      v8f c[4] = {};
#pragma unroll
      for (int kt = 0; kt < 32; ++kt) {
        v2f a = ldA(sD2, H2, kt, ml, mh);
#pragma unroll
        for (int t = 0; t < 4; ++t)
          c[t] = wmma4(a, ldBs(sB2, 32, t, kt, lane), c[t]);
      }
#pragma unroll
      for (int t = 0; t < 4; ++t) {
        int j = t * 16 + ml;
        float sg = sSig[s * H_SZ + j];
#pragma unroll
        for (int r = 0; r < 8; ++r) {
          int m = m0 + r + mh * 8;
          float z = noise[(size_t)s * B_SZ * H_SZ + (size_t)m * H_SZ + j];
          float yn = y[t][r] - c[t][r] * h + sg * z;
          y[t][r] = yn;
          out[(size_t)m * S_SZ * H_SZ + (size_t)(s + 1) * H_SZ + j] = yn;
        }
      }
    }
    __syncthreads();
  }
}

extern "C" void kernel_launch(void* const* d_in, const int* in_sizes, int n_in,
                              void* d_out, int out_size, void* d_ws, size_t ws_size,
                              hipStream_t stream) {
  (void)in_sizes; (void)n_in; (void)out_size; (void)d_ws; (void)ws_size;
  const float* ts    = (const float*)d_in[0];
  const float* noise = (const float*)d_in[1];
  const float* Wp1   = (const float*)d_in[2];
  const float* bp1   = (const float*)d_in[3];
  const float* Wp2   = (const float*)d_in[4];
  const float* bp2   = (const float*)d_in[5];
  const float* Wp3   = (const float*)d_in[6];
  // d_in[7] = bp3 (no effect on the gradient)
  const float* Wd1   = (const float*)d_in[8];
  const float* bd1   = (const float*)d_in[9];
  const float* Wd2   = (const float*)d_in[10];
  const float* bd2   = (const float*)d_in[11];
  const float* mind  = (const float*)d_in[12];
  const float* maxd  = (const float*)d_in[13];
  float* out = (float*)d_out;

  constexpr size_t kSmemFloats =
      (size_t)17 * 8 * 64 + 32 * 4 * 64 + 16 * 8 * 64 + 32 * 4 * 64 +  // imgs
      H2 + H_SZ + H_SZ + S_SZ + (size_t)NSTEP * H_SZ +
      16 * HP + 16 * H2 + 16 * H_SZ + 16 * H2;
  const size_t smem = kSmemFloats * sizeof(float);   // ~285 KB of 320 KB/WGP

  sde_scan_kernel<<<B_SZ / 16, 32, smem, stream>>>(
      ts, noise, Wp1, bp1, Wp2, bp2, Wp3, Wd1, bd1, Wd2, bd2, mind, maxd, out);
}